// simpleFly_4294967296512
// MI455X (gfx1250) — compile-verified
//
#include <hip/hip_runtime.h>

#define PNn 3
#define KNn 50
#define MBONn 10
#define DANn 10

typedef float v2f __attribute__((ext_vector_type(2)));
typedef float v8f __attribute__((ext_vector_type(8)));

__device__ __forceinline__ float rl(float v, int srclane) {
  return __int_as_float(__builtin_amdgcn_readlane(__float_as_int(v), srclane));
}

// DPP8 selector immediate for xor-permute within groups of 8 lanes
#define DPP8_XOR(m)                                                        \
  (((0 ^ (m)) << 0) | ((1 ^ (m)) << 3) | ((2 ^ (m)) << 6) |                \
   ((3 ^ (m)) << 9) | ((4 ^ (m)) << 12) | ((5 ^ (m)) << 15) |              \
   ((6 ^ (m)) << 18) | ((7 ^ (m)) << 21))

template <int M>
__device__ __forceinline__ float xor_dpp8(float v) {
  return __int_as_float(
      __builtin_amdgcn_mov_dpp8(__float_as_int(v), DPP8_XOR(M)));
}
// xor-8 within a 16-lane row == row_ror:8 (DPP16 ctrl 0x128)
__device__ __forceinline__ float ror8(float v) {
  return __int_as_float(__builtin_amdgcn_update_dpp(
      0, __float_as_int(v), 0x128, 0xf, 0xf, true));
}
// sum over the 16 lanes of each half (bits 0..3), pure VALU DPP
__device__ __forceinline__ float halfsum16(float v) {
  v += xor_dpp8<1>(v);
  v += xor_dpp8<2>(v);
  v += xor_dpp8<4>(v);
  v += ror8(v);
  return v;
}

// One wave32 per batch element. Plastic 16x64 (m x k) state held in WMMA C/D
// layout: tile q (k-cols 16q..16q+15), VGPR j holds row m=j (lanes 0-15) and
// m=j+8 (lanes 16-31), col k = 16q + (lane&15).
__global__ __launch_bounds__(256) void fly_kernel(
    const float* __restrict__ odor,        // [T,B,3]
    const float* __restrict__ context,     // [T,B]
    const float* __restrict__ pn_weight,   // [B,3,50]
    const float* __restrict__ kn_bias,     // [50]
    const float* __restrict__ apl_kc_w,    // [50,1]
    const float* __restrict__ apl_bias,    // [1]
    const float* __restrict__ mbon_dan_w,  // [10,10] (d,m)
    const float* __restrict__ mbon_bias,   // [10]
    const float* __restrict__ dan_mbon_w,  // [10,10] (m,d)
    const float* __restrict__ dan_dan_w,   // [10,10] (d',d)
    const float* __restrict__ dan_ctx_w,   // [1,10]
    const float* __restrict__ dan_bias,    // [10]
    const float* __restrict__ dec_W,       // [2,10]
    const float* __restrict__ dec_b,       // [2]
    const float* __restrict__ kc_w0,       // [B,50,10]
    const float* __restrict__ wact0,       // [B,50,10]
    float* __restrict__ out,               // [T,B,2]
    int B, int T) {
  const int lane = threadIdx.x & 31;
  const int b = blockIdx.x * (blockDim.x >> 5) + (threadIdx.x >> 5);
  if (b >= B) return;  // wave-uniform

  const int kk = lane & 15;  // k within a tile column
  const int jm8 = (lane >> 4) * 8;

  // lambda-lane mapping for 10-dim state: m<8 -> lane m ; m=8,9 -> lanes 16,17
  const int mown = (lane < 8) ? lane : (lane - 8);
  const bool mvalid = (lane < 8) | (lane == 16) | (lane == 17);

  // ---- per-lane constants (k side), replicated across lane halves ----
  float pnw[4][PNn], knb[4], aplw[4], kmask[4];
#pragma unroll
  for (int q = 0; q < 4; ++q) {
    int k = 16 * q + kk;
    bool v = k < KNn;
    kmask[q] = v ? 1.f : 0.f;
    knb[q] = v ? kn_bias[k] : 0.f;
    aplw[q] = v ? apl_kc_w[k] : 0.f;
#pragma unroll
    for (int p = 0; p < PNn; ++p)
      pnw[q][p] =
          v ? pn_weight[(size_t)b * (PNn * KNn) + (size_t)p * KNn + k] : 0.f;
  }

  // ---- plastic state in WMMA C/D layout (transposed: rows m, cols k) ----
  v8f kcw[4], wacc[4];
#pragma unroll
  for (int q = 0; q < 4; ++q) {
#pragma unroll
    for (int j = 0; j < 8; ++j) {
      int m = j + jm8;
      int k = 16 * q + kk;
      bool v = (m < MBONn) && (k < KNn);
      size_t idx = (size_t)b * (KNn * MBONn) + (size_t)k * MBONn + m;
      kcw[q][j] = v ? kc_w0[idx] : 0.f;
      wacc[q][j] = v ? wact0[idx] : 0.f;
    }
  }

  // ---- 10x10 weight columns at lambda lanes ----
  float mdW[DANn], ddW[DANn], dmW[MBONn];
#pragma unroll
  for (int d = 0; d < DANn; ++d) {
    mdW[d] = mvalid ? mbon_dan_w[d * MBONn + mown] : 0.f;  // col m
    ddW[d] = mvalid ? dan_dan_w[d * DANn + mown] : 0.f;    // col d
  }
#pragma unroll
  for (int m = 0; m < MBONn; ++m)
    dmW[m] = mvalid ? dan_mbon_w[m * DANn + mown] : 0.f;  // col d
  float ctxw = mvalid ? dan_ctx_w[mown] : 0.f;
  float danb = mvalid ? dan_bias[mown] : 0.f;
  float mbonb = mvalid ? mbon_bias[mown] : 0.f;

  // uniform decoder weights (uniform addresses -> scalar loads)
  float dw0[MBONn], dw1[MBONn];
#pragma unroll
  for (int m = 0; m < MBONn; ++m) {
    dw0[m] = dec_W[m];
    dw1[m] = dec_W[MBONn + m];
  }
  const float db0 = dec_b[0], db1 = dec_b[1];
  const float aplb = apl_bias[0];

  // DT=0.5 TAU=1 RC=5 TAU_W=2
  const float dt_tau = 0.5f, one_m = 0.5f;
  const float alpha = 0.5f / 5.5f, one_ma = 1.f - alpha;
  const float dt_tauw = 0.25f, one_mw = 0.75f;
  const float DTc = 0.5f;

  float kc[4] = {0, 0, 0, 0}, lowkc[4] = {0, 0, 0, 0};
  float apl = 0.f, mbon = 0.f, dan = 0.f, lowdan = 0.f;

  for (int t = 0; t < T; ++t) {
    // wave-uniform inputs
    size_t ob = ((size_t)t * B + b) * PNn;
    float o0 = odor[ob], o1 = odor[ob + 1], o2 = odor[ob + 2];
    float ctx = context[(size_t)t * B + b];

    // --- Kenyon update (uses OLD apl) + APL pre-sum ---
    float kcn[4], s = 0.f;
#pragma unroll
    for (int q = 0; q < 4; ++q) {
      float pv = fmaf(o2, pnw[q][2], fmaf(o1, pnw[q][1], o0 * pnw[q][0]));
      float r = fmaxf(pv - apl + knb[q], 0.f) * kmask[q];
      kcn[q] = fmaf(kc[q], one_m, dt_tau * r);
      s = fmaf(kcn[q], aplw[q], s);
    }
    // reduce within each 16-lane half (halves hold identical kc) — DPP only
    s = halfsum16(s);
    apl = fmaf(apl, one_m, dt_tau * fmaxf(s + aplb, 0.f));

    // --- kc_value partials against OLD kc_w (C-layout local FMAs) ---
    float part[8];
#pragma unroll
    for (int j = 0; j < 8; ++j) {
      float acc = kcn[0] * kcw[0][j];
      acc = fmaf(kcn[1], kcw[1][j], acc);
      acc = fmaf(kcn[2], kcw[2][j], acc);
      acc = fmaf(kcn[3], kcw[3][j], acc);
      part[j] = halfsum16(acc);
    }
    // pick this lambda-lane's kc_value: j = lane&7 (valid at lanes 0-7,16,17)
    float kcv = part[0];
#pragma unroll
    for (int j = 1; j < 8; ++j) kcv = ((lane & 7) == j) ? part[j] : kcv;

    // --- broadcast OLD dan (v_readlane from lambda lanes) ---
    float danS[DANn];
#pragma unroll
    for (int d = 0; d < DANn; ++d) danS[d] = rl(dan, d < 8 ? d : 8 + d);

    // --- MBON update ---
    float dmp = 0.f;
#pragma unroll
    for (int d = 0; d < DANn; ++d) dmp = fmaf(danS[d], mdW[d], dmp);
    float dan_mod = 1.f / (1.f + __expf(-dmp));
    float mbonn = fmaf(mbon, one_m, dt_tau * dan_mod * fmaxf(kcv + mbonb, 0.f));

    // broadcast NEW mbon
    float mbS[MBONn];
#pragma unroll
    for (int m = 0; m < MBONn; ++m) mbS[m] = rl(mbonn, m < 8 ? m : 8 + m);

    // --- DAN update (OLD dan, NEW mbon) ---
    float dp = fmaf(ctx, ctxw, danb);
#pragma unroll
    for (int d = 0; d < DANn; ++d) dp = fmaf(danS[d], ddW[d], dp);
#pragma unroll
    for (int m = 0; m < MBONn; ++m) dp = fmaf(mbS[m], dmW[m], dp);
    float dann = fmaf(dan, one_m, dt_tau * fmaxf(dp, 0.f));

    // --- low-pass traces ---
    float lowkcn[4];
#pragma unroll
    for (int q = 0; q < 4; ++q) lowkcn[q] = fmaf(alpha, kcn[q], one_ma * lowkc[q]);
    float lowdann = fmaf(alpha, dann, one_ma * lowdan);

    // --- kc_w blend toward OLD wact (before WMMA overwrites it) ---
#pragma unroll
    for (int q = 0; q < 4; ++q)
#pragma unroll
      for (int j = 0; j < 8; ++j)
        kcw[q][j] = fmaf(one_mw, kcw[q][j], dt_tauw * wacc[q][j]);

    // --- A operand: rows m at lanes 0-15 (m=8,9 via readlane from 16,17) ---
    float ld16 = rl(lowdann, 16), ld17 = rl(lowdann, 17);
    float dn16 = rl(dann, 16), dn17 = rl(dann, 17);
    bool lo8 = lane < 8;
    float aLD = lo8 ? lowdann : (lane == 8 ? ld16 : (lane == 9 ? ld17 : 0.f));
    float aDN = lo8 ? dann : (lane == 8 ? dn16 : (lane == 9 ? dn17 : 0.f));
    v2f A;
    A[0] = DTc * aLD;   // K=0 col (lanes 16-31 are K=2 col -> 0)
    A[1] = -DTc * aDN;  // K=1 col (lanes 16-31 are K=3 col -> 0)

    // --- wact += DT*(kc x low_dan - low_kc x dan), 4 WMMAs ---
#pragma unroll
    for (int q = 0; q < 4; ++q) {
      v2f Bm;
      Bm[0] = kcn[q];     // K=0 row; K=2 row slot irrelevant (A col 2 == 0)
      Bm[1] = lowkcn[q];  // K=1 row
      wacc[q] = __builtin_amdgcn_wmma_f32_16x16x4_f32(
          false, A, false, Bm, (short)0, wacc[q], false, false);
    }

    // --- decoder output ---
    float oo0 = db0, oo1 = db1;
#pragma unroll
    for (int m = 0; m < MBONn; ++m) {
      oo0 = fmaf(mbS[m], dw0[m], oo0);
      oo1 = fmaf(mbS[m], dw1[m], oo1);
    }
    if (lane == 0) {
      float2 ov = make_float2(oo0, oo1);
      *(float2*)(out + ((size_t)t * B + b) * 2) = ov;
    }

    // commit state
#pragma unroll
    for (int q = 0; q < 4; ++q) {
      kc[q] = kcn[q];
      lowkc[q] = lowkcn[q];
    }
    mbon = mbonn;
    dan = dann;
    lowdan = lowdann;
  }
}

extern "C" void kernel_launch(void* const* d_in, const int* in_sizes, int n_in,
                              void* d_out, int out_size, void* d_ws,
                              size_t ws_size, hipStream_t stream) {
  (void)n_in;
  (void)d_ws;
  (void)ws_size;
  (void)out_size;
  const float* odor = (const float*)d_in[0];
  const float* context = (const float*)d_in[1];
  const float* pn_weight = (const float*)d_in[2];
  const float* kn_bias = (const float*)d_in[3];
  const float* apl_kc_w = (const float*)d_in[4];
  const float* apl_bias = (const float*)d_in[5];
  const float* mbon_dan_w = (const float*)d_in[6];
  const float* mbon_bias = (const float*)d_in[7];
  const float* dan_mbon_w = (const float*)d_in[8];
  const float* dan_dan_w = (const float*)d_in[9];
  const float* dan_ctx_w = (const float*)d_in[10];
  const float* dan_bias = (const float*)d_in[11];
  const float* dec_W = (const float*)d_in[12];
  const float* dec_b = (const float*)d_in[13];
  const float* kc_w0 = (const float*)d_in[14];
  const float* wact0 = (const float*)d_in[15];

  int B = in_sizes[14] / (KNn * MBONn);
  int T = in_sizes[0] / (B * PNn);

  const int waves_per_block = 8;  // 256 threads = 8 wave32s
  int grid = (B + waves_per_block - 1) / waves_per_block;
  fly_kernel<<<grid, 32 * waves_per_block, 0, stream>>>(
      odor, context, pn_weight, kn_bias, apl_kc_w, apl_bias, mbon_dan_w,
      mbon_bias, dan_mbon_w, dan_dan_w, dan_ctx_w, dan_bias, dec_W, dec_b,
      kc_w0, wact0, (float*)d_out, B, T);
}